// MultiHeadAttentionPooling_77481210020095
// MI455X (gfx1250) — compile-verified
//
#include <hip/hip_runtime.h>
#include <math.h>

typedef __attribute__((ext_vector_type(2))) float v2f;
typedef __attribute__((ext_vector_type(8))) float v8f;

#define D     512
#define H     8
#define HD    64
#define BATCH 16
#define TTOT  12032          // sum(512 + 32*i, i=0..15)
#define SCALE 0.125f         // 1/sqrt(64)

// workspace layout (float offsets)
#define WS_MT  0             // 16*512   mT[h][d] (heads padded to 16, h>=8 zero)
#define WS_V0  8192          // 512      v0 = Wv*c + bv (class-token V, same for all b)
#define WS_SB  8704          // 16       sb[h] = q0_h . bk_h
#define WS_S0  8720          // 16       s0[h] = scale * q0_h . k0_h (class-token score)
#define WS_A0  8736          // 128      attn0[b*8+h] (class-token attention weight)
#define WS_CTX 8960          // 16*512
#define WS_U   17408         // 16*8*512 weighted sums
#define WS_SC  82944         // TTOT*8   scores -> normalized weights (in place)

// ---------------------------------------------------------------- setup -----
__global__ __launch_bounds__(256) void k_setup(const float* __restrict__ c,
                                               const float* __restrict__ W,
                                               const float* __restrict__ bI,
                                               float* __restrict__ ws) {
  __shared__ float q0[D];
  __shared__ float k0[D];
  const int tid = threadIdx.x;
  for (int r = tid; r < D; r += 256) {
    float aq = bI[r], ak = bI[D + r], av = bI[2 * D + r];
    const float* wq = W + (size_t)r * D;
    const float* wk = W + (size_t)(D + r) * D;
    const float* wv = W + (size_t)(2 * D + r) * D;
    for (int d = 0; d < D; ++d) {
      const float cv = c[d];
      aq += cv * wq[d];
      ak += cv * wk[d];
      av += cv * wv[d];
    }
    q0[r] = aq;
    k0[r] = ak;
    ws[WS_V0 + r] = av;
  }
  __syncthreads();
  // mT[h][d] = sum_e q0[h*64+e] * Wk[h*64+e][d]; heads 8..15 are zero padding
  for (int idx = tid; idx < 16 * D; idx += 256) {
    const int h = idx >> 9;
    const int d = idx & (D - 1);
    float acc = 0.f;
    if (h < H) {
      const float* wk = W + (size_t)(D + h * HD) * D + d;
      for (int e = 0; e < HD; ++e) acc += q0[h * HD + e] * wk[(size_t)e * D];
    }
    ws[WS_MT + idx] = acc;
  }
  if (tid < H) {
    float sb = 0.f, s0 = 0.f;
    for (int e = 0; e < HD; ++e) {
      const float q = q0[tid * HD + e];
      sb += q * bI[D + tid * HD + e];
      s0 += q * k0[tid * HD + e];
    }
    ws[WS_SB + tid] = sb;
    ws[WS_S0 + tid] = SCALE * s0;
  }
}

// --------------------------------------------- scores: T x 8 via f32 WMMA ---
// One wave per 16-token tile. A = x tile (16xK), B = mT (Kx16, heads padded).
// 16x16x4 f32 A layout: lanes 0-15 hold M=lane, K={0,1} in v0/v1; lanes 16-31
// hold K={2,3}. B mirrors with N=lane&15. Both become one b64 load per step.
__global__ __launch_bounds__(256) void k_scores(const float* __restrict__ x,
                                                float* __restrict__ ws) {
  const float* mT = ws + WS_MT;
  const float* sb = ws + WS_SB;
  float* sc = ws + WS_SC;
  const int wave = blockIdx.x * (blockDim.x >> 5) + (threadIdx.x >> 5);
  const int lane = threadIdx.x & 31;
  const int row = lane & 15;
  const int kh = lane >> 4;
  const int t0 = wave * 16;
  const float* ax = x + (size_t)(t0 + row) * D + 2 * kh;
  const float* bx = mT + (size_t)row * D + 2 * kh;
  v8f acc = {};
#pragma unroll 4
  for (int k = 0; k < D; k += 4) {
    v2f a = *(const v2f*)(ax + k);
    v2f b = *(const v2f*)(bx + k);
    acc = __builtin_amdgcn_wmma_f32_16x16x4_f32(false, a, false, b,
                                                (short)0, acc, false, false);
  }
  const int h = row;  // C/D layout: lane&15 = N (head), vgpr r -> M = r + 8*kh
  if (h < H) {
    const float sbh = sb[h];
#pragma unroll
    for (int r = 0; r < 8; ++r) {
      const int t = t0 + r + 8 * kh;
      sc[(size_t)t * H + h] = SCALE * (acc[r] + sbh);
    }
  }
}

// ------------------------------------------------- softmax per (batch,head) -
__global__ __launch_bounds__(256) void k_softmax(const int* __restrict__ lengths,
                                                 float* __restrict__ ws) {
  __shared__ float red[256];
  const int b = blockIdx.x >> 3;
  const int h = blockIdx.x & 7;
  const int tid = threadIdx.x;
  int off = 0;
  for (int i = 0; i < b; ++i) off += lengths[i];
  const int len = lengths[b];
  float* sc = ws + WS_SC + (size_t)off * H + h;
  const float s0 = ws[WS_S0 + h];

  float m = s0;
  for (int i = tid; i < len; i += 256) m = fmaxf(m, sc[(size_t)i * H]);
  red[tid] = m;
  __syncthreads();
  for (int s = 128; s > 0; s >>= 1) {
    if (tid < s) red[tid] = fmaxf(red[tid], red[tid + s]);
    __syncthreads();
  }
  m = red[0];
  __syncthreads();

  float sum = 0.f;
  for (int i = tid; i < len; i += 256) sum += expf(sc[(size_t)i * H] - m);
  red[tid] = sum;
  __syncthreads();
  for (int s = 128; s > 0; s >>= 1) {
    if (tid < s) red[tid] += red[tid + s];
    __syncthreads();
  }
  const float e0 = expf(s0 - m);
  const float Z = red[0] + e0;
  const float inv = 1.f / Z;
  if (tid == 0) ws[WS_A0 + b * H + h] = e0 * inv;
  for (int i = tid; i < len; i += 256)
    sc[(size_t)i * H] = expf(sc[(size_t)i * H] - m) * inv;
}

// ----------------------------------------- attn_weights = mean over heads ---
__global__ __launch_bounds__(256) void k_attnw(const float* __restrict__ ws,
                                               float* __restrict__ out) {
  const int t = blockIdx.x * 256 + threadIdx.x;
  if (t < TTOT) {
    const float* w = ws + WS_SC + (size_t)t * H;
    float s = 0.f;
#pragma unroll
    for (int h = 0; h < H; ++h) s += w[h];
    out[BATCH * D + t] = 0.125f * s;
  }
}

// ----------------------------- u[b,h,:] = sum_t w[t,h] * x[t,:] (per batch) -
__global__ __launch_bounds__(128) void k_u(const float* __restrict__ x,
                                           const int* __restrict__ lengths,
                                           float* __restrict__ ws) {
  __shared__ float wl[64 * H];
  const int b = blockIdx.x;
  const int d = blockIdx.y * 128 + threadIdx.x;
  int off = 0;
  for (int i = 0; i < b; ++i) off += lengths[i];
  const int len = lengths[b];
  const float* sc = ws + WS_SC + (size_t)off * H;
  float acc[H];
#pragma unroll
  for (int h = 0; h < H; ++h) acc[h] = 0.f;
  for (int c0 = 0; c0 < len; c0 += 64) {
    const int cnt = min(64, len - c0);
    __syncthreads();
    for (int j = threadIdx.x; j < cnt * H; j += 128) wl[j] = sc[(size_t)c0 * H + j];
    __syncthreads();
    for (int i = 0; i < cnt; ++i) {
      const float xv = x[(size_t)(off + c0 + i) * D + d];
#pragma unroll
      for (int h = 0; h < H; ++h) acc[h] += wl[i * H + h] * xv;
    }
  }
#pragma unroll
  for (int h = 0; h < H; ++h) ws[WS_U + (size_t)(b * H + h) * D + d] = acc[h];
}

// ------------- ctx[b,i] = Wv[i,:].u[b,head(i),:] + (1-a0)*bv[i] + a0*v0[i] --
__global__ __launch_bounds__(256) void k_ctx(const float* __restrict__ W,
                                             const float* __restrict__ bI,
                                             float* __restrict__ ws) {
  const int gid = blockIdx.x * 256 + threadIdx.x;  // 0..8191
  const int b = gid >> 9;
  const int i = gid & 511;
  const int h = i >> 6;
  const float* wr = W + (size_t)(2 * D + i) * D;
  const float* u = ws + WS_U + (size_t)(b * H + h) * D;
  float acc = 0.f;
  for (int d = 0; d < D; ++d) acc += wr[d] * u[d];
  const float a0 = ws[WS_A0 + b * H + h];
  ws[WS_CTX + gid] = acc + (1.f - a0) * bI[2 * D + i] + a0 * ws[WS_V0 + i];
}

// ---------------------------------- pooled[b,i] = Wo[i,:].ctx[b,:] + bo[i] --
__global__ __launch_bounds__(256) void k_out(const float* __restrict__ Wo,
                                             const float* __restrict__ bo,
                                             const float* __restrict__ ws,
                                             float* __restrict__ out) {
  const int gid = blockIdx.x * 256 + threadIdx.x;  // 0..8191
  const int b = gid >> 9;
  const int i = gid & 511;
  const float* wr = Wo + (size_t)i * D;
  const float* cx = ws + WS_CTX + (size_t)b * D;
  float acc = bo[i];
  for (int d = 0; d < D; ++d) acc += wr[d] * cx[d];
  out[gid] = acc;
}

extern "C" void kernel_launch(void* const* d_in, const int* in_sizes, int n_in,
                              void* d_out, int out_size, void* d_ws, size_t ws_size,
                              hipStream_t stream) {
  const float* x   = (const float*)d_in[0];
  const int* lens  = (const int*)d_in[1];
  const float* ct  = (const float*)d_in[2];
  const float* Wi  = (const float*)d_in[3];
  const float* bi  = (const float*)d_in[4];
  const float* Wo  = (const float*)d_in[5];
  const float* bo  = (const float*)d_in[6];
  float* out = (float*)d_out;
  float* ws  = (float*)d_ws;

  k_setup<<<1, 256, 0, stream>>>(ct, Wi, bi, ws);
  k_scores<<<TTOT / 16 / 8, 256, 0, stream>>>(x, ws);   // 752 waves, 94 blocks
  k_softmax<<<BATCH * H, 256, 0, stream>>>(lens, ws);
  k_attnw<<<TTOT / 256, 256, 0, stream>>>(ws, out);
  k_u<<<dim3(BATCH, D / 128), 128, 0, stream>>>(x, lens, ws);
  k_ctx<<<(BATCH * D) / 256, 256, 0, stream>>>(Wi, bi, ws);
  k_out<<<(BATCH * D) / 256, 256, 0, stream>>>(Wo, bo, ws, out);
}